// Model_48816598286780
// MI455X (gfx1250) — compile-verified
//
#include <hip/hip_runtime.h>
#include <math.h>

#define Bv 8
#define Cc 128
#define Nn 256
#define Tt 64
#define Hh 8

typedef __attribute__((ext_vector_type(2))) float v2f;
typedef __attribute__((ext_vector_type(8))) float v8f;
typedef __attribute__((ext_vector_type(4))) int v4i;

__device__ __forceinline__ v8f wmma4(v2f a, v2f b, v8f c) {
  // V_WMMA_F32_16X16X4_F32 : D = A(16x4) * B(4x16) + C(16x16)
  return __builtin_amdgcn_wmma_f32_16x16x4_f32(false, a, false, b, (short)0, c, false, false);
}

// ---------------------------------------------------------------------------
// Async global -> LDS staging (gfx1250 ASYNCcnt path), guarded fallback
// ---------------------------------------------------------------------------
#if defined(__has_builtin)
#if __has_builtin(__builtin_amdgcn_global_load_async_to_lds_b128)
#define ASYNC_LDS 1
#endif
#endif

__device__ __forceinline__ void async_copy16(float* lds, const float* g) {
#ifdef ASYNC_LDS
  __builtin_amdgcn_global_load_async_to_lds_b128(
      (__attribute__((address_space(1))) v4i*)g,
      (__attribute__((address_space(3))) v4i*)lds, 0, 0);
#else
  *(float4*)lds = *(const float4*)g;
#endif
}

__device__ __forceinline__ void async_wait() {
#ifdef ASYNC_LDS
#if defined(__has_builtin) && __has_builtin(__builtin_amdgcn_s_wait_asynccnt)
  __builtin_amdgcn_s_wait_asynccnt(0);
#else
  asm volatile("s_wait_asynccnt 0" ::: "memory");
#endif
#endif
}

// ---------------------------------------------------------------------------
// Laplacian helpers
// ---------------------------------------------------------------------------
__global__ void k_deg(const float* __restrict__ adj, float* __restrict__ dv) {
  int i = threadIdx.x;  // 256 threads, 1 block
  float s = 0.f;
  for (int j = 0; j < Nn; ++j) s += adj[i * Nn + j];
  dv[i] = rsqrtf(s);
}

__global__ void k_lap(const float* __restrict__ adj, const float* __restrict__ dv,
                      float* __restrict__ Lap) {
  int idx = blockIdx.x * 256 + threadIdx.x;
  int i = idx >> 8, j = idx & 255;
  Lap[idx] = adj[idx] * dv[i] * dv[j];
}

// ---------------------------------------------------------------------------
// Conv weight reformat: (O, I, 1, 3) -> [o][dt][i] (i-contiguous)
// 5 weights, each C*C*3 = 49152 elements
// ---------------------------------------------------------------------------
__global__ void k_wrefmt(const float* __restrict__ wq, const float* __restrict__ wk,
                         const float* __restrict__ dxc, const float* __restrict__ dec,
                         const float* __restrict__ dc1, float* __restrict__ out) {
  int idx = blockIdx.x * 256 + threadIdx.x;  // 5*49152 = 245760 total
  int wsel = idx / 49152, r = idx % 49152;
  int o = r / 384, rem = r % 384, dt = rem >> 7, i = rem & 127;
  const float* src = (wsel == 0) ? wq : (wsel == 1) ? wk : (wsel == 2) ? dxc
                                                     : (wsel == 3)   ? dec
                                                                     : dc1;
  out[idx] = src[(o * Cc + i) * 3 + dt];
}

// ---------------------------------------------------------------------------
// Diffusion step embedding
// ---------------------------------------------------------------------------
__global__ void k_emb(const int* __restrict__ step,
                      const float* __restrict__ p1w, const float* __restrict__ p1b,
                      const float* __restrict__ p2w, const float* __restrict__ p2b,
                      float* __restrict__ emb) {
  __shared__ float tab[Cc];
  __shared__ float e1[Cc];
  int b = blockIdx.x, tid = threadIdx.x;  // 128 threads
  float sv = (float)step[b];
  if (tid < 64) {
    float f = powf(10.f, ((float)tid / 63.f) * 4.f);
    float arg = sv * f;
    tab[tid] = sinf(arg);
    tab[tid + 64] = cosf(arg);
  }
  __syncthreads();
  float a = p1b[tid];
  for (int i = 0; i < Cc; ++i) a += p1w[tid * Cc + i] * tab[i];
  a = a / (1.f + expf(-a));
  e1[tid] = a;
  __syncthreads();
  float a2 = p2b[tid];
  for (int i = 0; i < Cc; ++i) a2 += p2w[tid * Cc + i] * e1[i];
  a2 = a2 / (1.f + expf(-a2));
  emb[b * Cc + tid] = a2;
}

// ---------------------------------------------------------------------------
// Temporal attention mega-kernel: one block per (b,n)
// LDS: Xt [t_pad][c] s132 (later OL [t][c] s130), Qct/Kct [t][c] s132
//      (Qct later OUT2 s130), Vl [c][t] s66, Pst per-wave 16x18
// ---------------------------------------------------------------------------
__global__ void __launch_bounds__(256) k_temporal(
    const float* __restrict__ x,
    const float* __restrict__ wqr, const float* __restrict__ bq,
    const float* __restrict__ wkr, const float* __restrict__ bk,
    const float* __restrict__ w1, const float* __restrict__ b1,
    const float* __restrict__ w2, const float* __restrict__ b2v,
    const float* __restrict__ ln0g, const float* __restrict__ ln0b,
    float* __restrict__ xp_out) {
  extern __shared__ float sm[];
  float* Xt = sm;                   // 66 x 132 = 8712 (later OL 64 x 130)
  float* Qct = sm + 8712;           // 64 x 132 = 8448 (later OUT2 64 x 130)
  float* Kct = sm + 17160;          // 64 x 132 = 8448
  float* Vl = sm + 25608;           // 128 x 66 = 8448
  float* Pst = sm + 34056;          // 8 * 288 = 2304
  float* mu = sm + 36360;           // 64
  float* rsd = sm + 36424;          // 64

  int tid = threadIdx.x, lane = tid & 31, w = tid >> 5;
  int lr = lane & 15, hf = lane >> 4, kk = hf * 2;
  int bn = blockIdx.x;
  int b = bn >> 8, n = bn & 255;

  // phase a: load X (C x T), transpose into Xt[t+1][c], zero-pad t rows
  for (int c = tid; c < 132; c += 256) { Xt[c] = 0.f; Xt[65 * 132 + c] = 0.f; }
  const float* xb = x + (size_t)b * Cc * Nn * Tt + (size_t)n * Tt;  // + c*(N*T) + t
  for (int q = tid; q < Cc * (Tt / 4); q += 256) {
    int c = q >> 4, t4 = (q & 15) * 4;
    float4 xv = *(const float4*)&xb[(size_t)c * Nn * Tt + t4];
    Xt[(t4 + 1) * 132 + c] = xv.x;
    Xt[(t4 + 2) * 132 + c] = xv.y;
    Xt[(t4 + 3) * 132 + c] = xv.z;
    Xt[(t4 + 4) * 132 + c] = xv.w;
  }
  __syncthreads();

  int row0 = w * 16;
  // phase b1: fused Q/K conv (shared B fragments)
  {
    v8f aq[4] = {}, ak[4] = {};
    for (int dt = 0; dt < 3; ++dt)
      for (int k = 0; k < 32; ++k) {
        int i0 = k * 4 + kk;
        v2f a1 = *(const v2f*)&wqr[((row0 + lr) * 3 + dt) * Cc + i0];
        v2f a2 = *(const v2f*)&wkr[((row0 + lr) * 3 + dt) * Cc + i0];
#pragma unroll
        for (int nt = 0; nt < 4; ++nt) {
          v2f bf = *(const v2f*)&Xt[(nt * 16 + lr + dt) * 132 + i0];
          aq[nt] = wmma4(a1, bf, aq[nt]);
          ak[nt] = wmma4(a2, bf, ak[nt]);
        }
      }
#pragma unroll
    for (int v = 0; v < 8; ++v) {
      int row = row0 + v + hf * 8;
      float bb1 = bq[row], bb2 = bk[row];
#pragma unroll
      for (int nt = 0; nt < 4; ++nt) {
        int t = nt * 16 + lr;
        Qct[t * 132 + row] = aq[nt][v] + bb1;
        Kct[t * 132 + row] = ak[nt][v] + bb2;
      }
    }
  }
  // phase b2: V = w1 @ X + b1, stored [c][t] stride 66
  {
    v8f av[4] = {};
    for (int k = 0; k < 32; ++k) {
      int i0 = k * 4 + kk;
      v2f a = *(const v2f*)&w1[(row0 + lr) * Cc + i0];
#pragma unroll
      for (int nt = 0; nt < 4; ++nt) {
        v2f bf = *(const v2f*)&Xt[(nt * 16 + lr + 1) * 132 + i0];
        av[nt] = wmma4(a, bf, av[nt]);
      }
    }
#pragma unroll
    for (int v = 0; v < 8; ++v) {
      int row = row0 + v + hf * 8;
      float bb = b1[row];
#pragma unroll
      for (int nt = 0; nt < 4; ++nt)
        Vl[row * 66 + nt * 16 + lr] = av[nt][v] + bb;
    }
  }
  __syncthreads();

  // phase c: attention, head h == wave id
  float* OL = Xt;             // 64 x 130
  float* Pw = Pst + w * 288;  // 16 x 18 per-wave scratch
  int h = w;
  for (int m = 0; m < 4; ++m) {
    v8f s[4] = {};
#pragma unroll
    for (int k = 0; k < 4; ++k) {
      int d0 = h * 16 + k * 4 + kk;
      v2f a = *(const v2f*)&Qct[(m * 16 + lr) * 132 + d0];
#pragma unroll
      for (int nt = 0; nt < 4; ++nt) {
        v2f bf = *(const v2f*)&Kct[(nt * 16 + lr) * 132 + d0];
        s[nt] = wmma4(a, bf, s[nt]);
      }
    }
    // softmax over 64 cols, scale 1/sqrt(16)=0.25
#pragma unroll
    for (int v = 0; v < 8; ++v) {
      float mx = fmaxf(fmaxf(s[0][v], s[1][v]), fmaxf(s[2][v], s[3][v]));
      for (int o = 1; o < 16; o <<= 1) mx = fmaxf(mx, __shfl_xor(mx, o, 32));
      float sum = 0.f;
#pragma unroll
      for (int nt = 0; nt < 4; ++nt) {
        float ev = __expf((s[nt][v] - mx) * 0.25f);
        s[nt][v] = ev;
        sum += ev;
      }
      for (int o = 1; o < 16; o <<= 1) sum += __shfl_xor(sum, o, 32);
      float inv = 1.f / sum;
#pragma unroll
      for (int nt = 0; nt < 4; ++nt) s[nt][v] *= inv;
    }
    // O = P @ V_h, per 16-col K chunk via LDS transpose of P
    v8f oacc = {};
#pragma unroll
    for (int nt = 0; nt < 4; ++nt) {
#pragma unroll
      for (int v = 0; v < 8; ++v) Pw[(v + hf * 8) * 18 + lr] = s[nt][v];
#pragma unroll
      for (int k4 = 0; k4 < 4; ++k4) {
        v2f a = *(const v2f*)&Pw[lr * 18 + k4 * 4 + kk];
        int srow = nt * 16 + k4 * 4 + kk;
        v2f bf = *(const v2f*)&Vl[(h * 16 + lr) * 66 + srow];
        oacc = wmma4(a, bf, oacc);
      }
    }
#pragma unroll
    for (int v = 0; v < 8; ++v)
      OL[(m * 16 + v + hf * 8) * 130 + h * 16 + lr] = oacc[v];
  }
  __syncthreads();

  // phase d: OUT2 = O @ w2^T + b2 ; wave w owns output cols [16w,16w+16)
  float* OUT2 = Qct;  // 64 x 130
  {
    v8f acc[4] = {};
    for (int k = 0; k < 32; ++k) {
      int i0 = k * 4 + kk;
      v2f bf = *(const v2f*)&w2[(w * 16 + lr) * Cc + i0];
#pragma unroll
      for (int m = 0; m < 4; ++m) {
        v2f a = *(const v2f*)&OL[(m * 16 + lr) * 130 + i0];
        acc[m] = wmma4(a, bf, acc[m]);
      }
    }
    float bb = b2v[w * 16 + lr];
#pragma unroll
    for (int m = 0; m < 4; ++m)
#pragma unroll
      for (int v = 0; v < 8; ++v)
        OUT2[(m * 16 + v + hf * 8) * 130 + w * 16 + lr] = acc[m][v] + bb;
  }
  __syncthreads();

  // phase e: LayerNorm over C + residual x, store xp (B,N,T,C)
  if (tid < Tt) {
    float s1 = 0.f, s2 = 0.f;
    for (int c = 0; c < Cc; ++c) {
      float z = OUT2[tid * 130 + c];
      s1 += z;
      s2 += z * z;
    }
    float m_ = s1 / (float)Cc;
    float v_ = s2 / (float)Cc - m_ * m_;
    mu[tid] = m_;
    rsd[tid] = rsqrtf(v_ + 1e-5f);
  }
  __syncthreads();
  float* xpo = xp_out + (size_t)bn * Tt * Cc;
  for (int idx = tid; idx < Tt * Cc; idx += 256) {
    int t = idx >> 7, c = idx & 127;
    float z = (OUT2[t * 130 + c] - mu[t]) * rsd[t] * ln0g[c] + ln0b[c];
    xpo[idx] = xb[(size_t)c * Nn * Tt + t] + z;
  }
}

// ---------------------------------------------------------------------------
// Spatial attention + GCN mega-kernel: one block per (b,t)
// ---------------------------------------------------------------------------
__global__ void __launch_bounds__(256) k_spatial(
    const float* __restrict__ xp, const float* __restrict__ Lap,
    const float* __restrict__ theta,
    const float* __restrict__ ln1g, const float* __restrict__ ln1b,
    const float* __restrict__ emb, float* __restrict__ h_out) {
  extern __shared__ float sm[];
  float* Xs = sm;           // 256 x 132 = 33792
  float* Pst = sm + 33792;  // 8 * 288
  int tid = threadIdx.x, lane = tid & 31, w = tid >> 5;
  int lr = lane & 15, hf = lane >> 4, kk = hf * 2;
  int bt = blockIdx.x;
  int b = bt >> 6, t = bt & 63;

  __builtin_prefetch(theta + tid, 0, 1);

  const float* xpb = xp + ((size_t)b * Nn * Tt + t) * Cc;  // + n*T*C + c
  for (int q = tid; q < Nn * (Cc / 4); q += 256) {
    int n = q >> 5, c4 = (q & 31) * 4;
    async_copy16(&Xs[n * 132 + c4], &xpb[(size_t)n * Tt * Cc + c4]);
  }
  async_wait();
  __syncthreads();

  const float sc = 0.0883883476483184f;  // 1/sqrt(128)
  float* Pw = Pst + w * 288;

  for (int mi = 0; mi < 2; ++mi) {
    int mm = w + mi * 8;  // 16 m-tiles of 16 nodes
    // S row-block (16 x 256) = Xs_rows @ Xs^T
    v8f s[16] = {};
    for (int k = 0; k < 32; ++k) {
      int i0 = k * 4 + kk;
      v2f a = *(const v2f*)&Xs[(mm * 16 + lr) * 132 + i0];
#pragma unroll
      for (int nt = 0; nt < 16; ++nt) {
        v2f bf = *(const v2f*)&Xs[(nt * 16 + lr) * 132 + i0];
        s[nt] = wmma4(a, bf, s[nt]);
      }
    }
    // softmax(S*sc) * sc * Lap  (rows of 256)
#pragma unroll
    for (int v = 0; v < 8; ++v) {
      float mx = -1e30f;
#pragma unroll
      for (int nt = 0; nt < 16; ++nt) mx = fmaxf(mx, s[nt][v]);
      for (int o = 1; o < 16; o <<= 1) mx = fmaxf(mx, __shfl_xor(mx, o, 32));
      float sum = 0.f;
#pragma unroll
      for (int nt = 0; nt < 16; ++nt) {
        float ev = __expf((s[nt][v] - mx) * sc);
        s[nt][v] = ev;
        sum += ev;
      }
      for (int o = 1; o < 16; o <<= 1) sum += __shfl_xor(sum, o, 32);
      float inv = sc / sum;  // outer *sc folded into normalization
      int rown = mm * 16 + v + hf * 8;
#pragma unroll
      for (int nt = 0; nt < 16; ++nt)
        s[nt][v] *= inv * Lap[rown * Nn + nt * 16 + lr];
    }
    // Y (16 x 128) = G' @ Xs
    v8f y[8] = {};
#pragma unroll
    for (int nt = 0; nt < 16; ++nt) {
#pragma unroll
      for (int v = 0; v < 8; ++v) Pw[(v + hf * 8) * 18 + lr] = s[nt][v];
#pragma unroll
      for (int k4 = 0; k4 < 4; ++k4) {
        v2f a = *(const v2f*)&Pw[lr * 18 + k4 * 4 + kk];
        int r0 = nt * 16 + k4 * 4 + kk;
#pragma unroll
        for (int ct = 0; ct < 8; ++ct) {
          v2f bf;
          bf.x = Xs[r0 * 132 + ct * 16 + lr];
          bf.y = Xs[(r0 + 1) * 132 + ct * 16 + lr];
          y[ct] = wmma4(a, bf, y[ct]);
        }
      }
    }
    // Z (16 x 128) = relu(Y @ theta^T)
    v8f z[8] = {};
#pragma unroll
    for (int kc = 0; kc < 8; ++kc) {
#pragma unroll
      for (int v = 0; v < 8; ++v) Pw[(v + hf * 8) * 18 + lr] = y[kc][v];
#pragma unroll
      for (int k4 = 0; k4 < 4; ++k4) {
        v2f a = *(const v2f*)&Pw[lr * 18 + k4 * 4 + kk];
        int i0 = kc * 16 + k4 * 4 + kk;
#pragma unroll
        for (int ot = 0; ot < 8; ++ot) {
          v2f bf = *(const v2f*)&theta[(ot * 16 + lr) * Cc + i0];
          z[ot] = wmma4(a, bf, z[ot]);
        }
      }
    }
#pragma unroll
    for (int ot = 0; ot < 8; ++ot)
#pragma unroll
      for (int v = 0; v < 8; ++v) z[ot][v] = fmaxf(z[ot][v], 0.f);

    // register LayerNorm over C + residual + diffusion embedding -> h
#pragma unroll
    for (int v = 0; v < 8; ++v) {
      float s1 = 0.f, s2 = 0.f;
#pragma unroll
      for (int ot = 0; ot < 8; ++ot) {
        float zz = z[ot][v];
        s1 += zz;
        s2 += zz * zz;
      }
      for (int o = 1; o < 16; o <<= 1) {
        s1 += __shfl_xor(s1, o, 32);
        s2 += __shfl_xor(s2, o, 32);
      }
      float m_ = s1 / (float)Cc;
      float var_ = s2 / (float)Cc - m_ * m_;
      float rs_ = rsqrtf(var_ + 1e-5f);
      int n = mm * 16 + v + hf * 8;
      const float* xr = xpb + (size_t)n * Tt * Cc;
      float* hr = h_out + ((size_t)(b * Nn + n) * Tt + t) * Cc;
#pragma unroll
      for (int ot = 0; ot < 8; ++ot) {
        int c = ot * 16 + lr;
        float val = (z[ot][v] - m_) * rs_ * ln1g[c] + ln1b[c];
        hr[c] = xr[c] + val + emb[b * Cc + c];
      }
    }
  }
}

// ---------------------------------------------------------------------------
// Decoder: one block per (b,n)
// ---------------------------------------------------------------------------
__global__ void __launch_bounds__(256) k_decoder(
    const float* __restrict__ h,
    const float* __restrict__ dxcr, const float* __restrict__ dxcb,
    const float* __restrict__ decr, const float* __restrict__ decb,
    const float* __restrict__ dc1r, const float* __restrict__ dc1b,
    const float* __restrict__ dp2w, const float* __restrict__ dp2b,
    float* __restrict__ out) {
  extern __shared__ float sm[];
  float* Hn = sm;          // 66 x 132 (time-padded, [t+1][c])
  float* Ec = sm + 8712;   // 66 x 132
  float* Ou = sm + 17424;  // 128 x 66
  int tid = threadIdx.x, lane = tid & 31, w = tid >> 5;
  int lr = lane & 15, hf = lane >> 4, kk = hf * 2;
  int bn = blockIdx.x;
  int b = bn >> 8, n = bn & 255;

  for (int c = tid; c < 132; c += 256) {
    Hn[c] = 0.f;
    Hn[65 * 132 + c] = 0.f;
    Ec[c] = 0.f;
    Ec[65 * 132 + c] = 0.f;
  }
  const float* hb = h + (size_t)bn * Tt * Cc;
  for (int q = tid; q < Tt * (Cc / 4); q += 256) {
    int t = q >> 5, c4 = (q & 31) * 4;
    async_copy16(&Hn[(t + 1) * 132 + c4], &hb[t * Cc + c4]);
  }
  async_wait();
  __syncthreads();

  int row0 = w * 16;
  // ec = conv(h, dec_w) + dec_b, staged to Ec[t+1][o]
  {
    v8f acc[4] = {};
    for (int dt = 0; dt < 3; ++dt)
      for (int k = 0; k < 32; ++k) {
        int i0 = k * 4 + kk;
        v2f a = *(const v2f*)&decr[((row0 + lr) * 3 + dt) * Cc + i0];
#pragma unroll
        for (int nt = 0; nt < 4; ++nt) {
          v2f bf = *(const v2f*)&Hn[(nt * 16 + lr + dt) * 132 + i0];
          acc[nt] = wmma4(a, bf, acc[nt]);
        }
      }
#pragma unroll
    for (int v = 0; v < 8; ++v) {
      int o = row0 + v + hf * 8;
      float bb = decb[o];
#pragma unroll
      for (int nt = 0; nt < 4; ++nt) {
        int t = nt * 16 + lr;
        Ec[(t + 1) * 132 + o] = acc[nt][v] + bb;
      }
    }
  }
  __syncthreads();

  // out = relu(conv(h,dxc) + sigmoid(ec) + conv(ec,dc1))
  {
    v8f ax[4] = {}, ag[4] = {};
    for (int dt = 0; dt < 3; ++dt)
      for (int k = 0; k < 32; ++k) {
        int i0 = k * 4 + kk;
        v2f a1 = *(const v2f*)&dxcr[((row0 + lr) * 3 + dt) * Cc + i0];
        v2f a2 = *(const v2f*)&dc1r[((row0 + lr) * 3 + dt) * Cc + i0];
#pragma unroll
        for (int nt = 0; nt < 4; ++nt) {
          int trow = nt * 16 + lr + dt;
          v2f b1f = *(const v2f*)&Hn[trow * 132 + i0];
          v2f b2f = *(const v2f*)&Ec[trow * 132 + i0];
          ax[nt] = wmma4(a1, b1f, ax[nt]);
          ag[nt] = wmma4(a2, b2f, ag[nt]);
        }
      }
#pragma unroll
    for (int v = 0; v < 8; ++v) {
      int o = row0 + v + hf * 8;
      float bx = dxcb[o], bg = dc1b[o];
#pragma unroll
      for (int nt = 0; nt < 4; ++nt) {
        int t = nt * 16 + lr;
        float ecv = Ec[(t + 1) * 132 + o];
        float val = (ax[nt][v] + bx) + 1.f / (1.f + __expf(-ecv)) + (ag[nt][v] + bg);
        Ou[o * 66 + t] = fmaxf(val, 0.f);
      }
    }
  }
  __syncthreads();

  // final 1x1 projection to 2 channels: out (B,2,N,T)
  if (tid < 2 * Tt) {
    int o2 = tid >> 6, t = tid & 63;
    float s = dp2b[o2];
    for (int c = 0; c < Cc; ++c) s += dp2w[o2 * Cc + c] * Ou[c * 66 + t];
    out[((size_t)(b * 2 + o2) * Nn + n) * Tt + t] = s;
  }
}

// ---------------------------------------------------------------------------
// Launch
// ---------------------------------------------------------------------------
extern "C" void kernel_launch(void* const* d_in, const int* in_sizes, int n_in,
                              void* d_out, int out_size, void* d_ws, size_t ws_size,
                              hipStream_t stream) {
  const float* x = (const float*)d_in[0];
  const float* adj = (const float*)d_in[1];
  const int* step = (const int*)d_in[2];
  const float* wq = (const float*)d_in[3];
  const float* bq = (const float*)d_in[4];
  const float* wk = (const float*)d_in[5];
  const float* bk = (const float*)d_in[6];
  const float* w1 = (const float*)d_in[7];
  const float* b1 = (const float*)d_in[8];
  const float* w2 = (const float*)d_in[9];
  const float* b2 = (const float*)d_in[10];
  const float* th = (const float*)d_in[11];
  const float* ln0g = (const float*)d_in[12];
  const float* ln0b = (const float*)d_in[13];
  const float* ln1g = (const float*)d_in[14];
  const float* ln1b = (const float*)d_in[15];
  const float* p1w = (const float*)d_in[16];
  const float* p1b = (const float*)d_in[17];
  const float* p2w = (const float*)d_in[18];
  const float* p2b = (const float*)d_in[19];
  const float* dxcw = (const float*)d_in[20];
  const float* dxcb = (const float*)d_in[21];
  const float* decw = (const float*)d_in[22];
  const float* decb = (const float*)d_in[23];
  const float* dc1w = (const float*)d_in[24];
  const float* dc1b = (const float*)d_in[25];
  const float* dp2w = (const float*)d_in[26];
  const float* dp2b = (const float*)d_in[27];

  float* ws = (float*)d_ws;
  float* Lap = ws;                     // 65536
  float* dv = ws + 65536;              // 256
  float* emb = ws + 65792;             // 1024
  float* wqr = ws + 66816;             // 49152
  float* wkr = ws + 115968;            // 49152
  float* dxcr = ws + 165120;           // 49152
  float* decr = ws + 214272;           // 49152
  float* dc1r = ws + 263424;           // 49152
  float* xp = ws + 312576;             // B*N*T*C = 16777216
  float* hbf = xp + (size_t)16777216;  // B*N*T*C

  k_deg<<<1, 256, 0, stream>>>(adj, dv);
  k_lap<<<Nn, 256, 0, stream>>>(adj, dv, Lap);
  k_wrefmt<<<960, 256, 0, stream>>>(wq, wk, dxcw, decw, dc1w, wqr);
  k_emb<<<Bv, 128, 0, stream>>>(step, p1w, p1b, p2w, p2b, emb);

  size_t sh_t = (size_t)36488 * sizeof(float);  // 145,952 B
  k_temporal<<<Bv * Nn, 256, sh_t, stream>>>(x, wqr, bq, wkr, bk, w1, b1, w2, b2,
                                             ln0g, ln0b, xp);

  size_t sh_s = (size_t)(33792 + 2304) * sizeof(float);  // 144,384 B
  k_spatial<<<Bv * Tt, 256, sh_s, stream>>>(xp, Lap, th, ln1g, ln1b, emb, hbf);

  size_t sh_d = (size_t)(8712 * 2 + 8448) * sizeof(float);  // 103,488 B
  k_decoder<<<Bv * Nn, 256, sh_d, stream>>>(hbf, dxcr, dxcb, decr, decb,
                                            dc1r, dc1b, dp2w, dp2b, (float*)d_out);
}